// DTWShpTime_80719615361844
// MI455X (gfx1250) — compile-verified
//
#include <hip/hip_runtime.h>
#include <math.h>
#include <stdint.h>

#define NT    128
#define RS    130              // R row stride (conflict-free: addr = i*129 + d)
#define GAMMA_F 0.01f
#define INVG_F  100.0f
#define BIGF    100000000.0f

// One workgroup per row. 128 threads = 4 wave32s. Thread tid owns DP row i = tid+1.
__global__ __launch_bounds__(128) void dtw_row_kernel(const float* __restrict__ input,
                                                      const float* __restrict__ target,
                                                      float* __restrict__ ws)
{
    __shared__ float Rm[(NT + 1) * RS];   // 129*130 floats = 67,080 B, fits CDNA5 WGP LDS
    __shared__ float tS[NT];
    __shared__ float pS[NT];
    __shared__ float eBuf[3][NT + 4];     // rotating E anti-diagonals
    __shared__ float red[NT];

    const int tid = threadIdx.x;
    const int row = blockIdx.x;
    const int b   = row >> 8;            // row / (C*H*W)
    const int chw = row & 255;
    const long base = (long)b * (NT * 256) + chw;   // element offset; time stride = 256

    // --- CDNA5 async global->LDS staging of the two length-128 sequences ---
    {
        const float* gp = input  + base + (long)tid * 256;
        const float* gt = target + base + (long)tid * 256;
        unsigned ldsP = (unsigned)(uintptr_t)(&pS[tid]);
        unsigned ldsT = (unsigned)(uintptr_t)(&tS[tid]);
        asm volatile("global_load_async_to_lds_b32 %0, %1, off" :: "v"(ldsP), "v"(gp) : "memory");
        asm volatile("global_load_async_to_lds_b32 %0, %1, off" :: "v"(ldsT), "v"(gt) : "memory");
        asm volatile("s_wait_asynccnt 0" ::: "memory");
    }

    // --- init R boundaries: R[0][0]=0, row0/col0 = BIG ---
    if (tid == 0) Rm[0] = 0.0f;
    Rm[tid + 1]        = BIGF;           // R[0][j], j=1..128
    Rm[(tid + 1) * RS] = BIGF;           // R[i][0], i=1..128
    __syncthreads();

    const int   i  = tid + 1;
    const float ti = tS[tid];            // t[i-1]

    // --- forward soft-DTW wavefront over anti-diagonals d = i+j ---
    for (int d = 2; d <= 2 * NT; ++d) {
        int j = d - i;
        if (j >= 1 && j <= NT) {
            float diff = ti - pS[j - 1];
            float Dij  = diff * diff;
            float a  = Rm[(i - 1) * RS + (j - 1)];
            float bb = Rm[(i - 1) * RS + j];
            float c  = Rm[i * RS + (j - 1)];
            float m  = fminf(a, fminf(bb, c));
            float s  = expf((m - a) * INVG_F) + expf((m - bb) * INVG_F) + expf((m - c) * INVG_F);
            Rm[i * RS + j] = Dij + (m - GAMMA_F * logf(s));
        }
        __syncthreads();
    }
    const float shape_r = Rm[NT * RS + NT];

    // --- backward pass: E recurrence on rotating diagonals; accumulate E*(i-j)^2 ---
    for (int k = tid; k < 3 * (NT + 4); k += NT) ((float*)eBuf)[k] = 0.0f;
    __syncthreads();
    if (tid == 0) eBuf[(2 * NT) % 3][NT] = 1.0f;   // E[N][N] = 1 (diag 2N), Omega contribution = 0
    __syncthreads();

    float acc = 0.0f;
    for (int d = 2 * NT - 1; d >= 2; --d) {
        float*       ec = eBuf[d % 3];
        const float* e1 = eBuf[(d + 1) % 3];       // diagonal d+1
        const float* e2 = eBuf[(d + 2) % 3];       // diagonal d+2
        int j = d - i;
        float Ev = 0.0f;
        if (j >= 1 && j <= NT) {
            float Rij = Rm[i * RS + j];
            if (i + 1 <= NT) {                     // child (i+1, j)
                float dd = tS[i] - pS[j - 1];
                Ev += expf((Rm[(i + 1) * RS + j] - Rij - dd * dd) * INVG_F) * e1[i + 1];
            }
            if (j + 1 <= NT) {                     // child (i, j+1)
                float dd = ti - pS[j];
                Ev += expf((Rm[i * RS + (j + 1)] - Rij - dd * dd) * INVG_F) * e1[i];
            }
            if (i + 1 <= NT && j + 1 <= NT) {      // child (i+1, j+1)
                float dd = tS[i] - pS[j];
                Ev += expf((Rm[(i + 1) * RS + (j + 1)] - Rij - dd * dd) * INVG_F) * e2[i + 1];
            }
            int dij = i - j;
            acc += Ev * (float)(dij * dij);
        }
        ec[i] = Ev;                                // clears stale diag d+3 slot too
        __syncthreads();
    }

    // --- deterministic block reduction of temporal accumulator ---
    red[tid] = acc;
    __syncthreads();
    for (int s = NT / 2; s > 0; s >>= 1) {
        if (tid < s) red[tid] += red[tid + s];
        __syncthreads();
    }
    if (tid == 0) {
        ws[row]        = shape_r;
        ws[1024 + row] = red[0] / (float)(NT * NT);
    }
}

__global__ __launch_bounds__(256) void dtw_final_kernel(const float* __restrict__ ws,
                                                        float* __restrict__ out)
{
    __shared__ float s1[256];
    __shared__ float s2[256];
    const int tid = threadIdx.x;
    float a = 0.0f, b = 0.0f;
    for (int k = tid; k < 1024; k += 256) {
        a += ws[k];
        b += ws[1024 + k];
    }
    s1[tid] = a; s2[tid] = b;
    __syncthreads();
    for (int s = 128; s > 0; s >>= 1) {
        if (tid < s) { s1[tid] += s1[tid + s]; s2[tid] += s2[tid + s]; }
        __syncthreads();
    }
    if (tid == 0) {
        float ls = s1[0] / 1024.0f;
        float lt = s2[0] / 1024.0f;
        out[0] = 0.5f * ls + 0.5f * lt;
        out[1] = ls;
        out[2] = lt;
    }
}

extern "C" void kernel_launch(void* const* d_in, const int* in_sizes, int n_in,
                              void* d_out, int out_size, void* d_ws, size_t ws_size,
                              hipStream_t stream) {
    const float* input  = (const float*)d_in[0];
    const float* target = (const float*)d_in[1];
    float* ws  = (float*)d_ws;          // 2048 floats: [0..1023]=shape, [1024..2047]=temporal
    float* out = (float*)d_out;         // 3 floats: loss, loss_shape, loss_temporal
    dtw_row_kernel<<<1024, 128, 0, stream>>>(input, target, ws);
    dtw_final_kernel<<<1, 256, 0, stream>>>(ws, out);
}